// Flow_80281528697438
// MI455X (gfx1250) — compile-verified
//
#include <hip/hip_runtime.h>
#include <math.h>

typedef __attribute__((ext_vector_type(2))) float v2f;
typedef __attribute__((ext_vector_type(8))) float v8f;

#define TRUNC_CHAMFER 2.0f
#define KSPLIT 4

// ---------------------------------------------------------------------------
// Pack kernel: build (c0, c1, c2, |c|^2) float4 arrays for the 4 point sets.
// ---------------------------------------------------------------------------
__global__ __launch_bounds__(256) void chamfer_pack(
    const float* __restrict__ fw, const float* __restrict__ bw,
    const float* __restrict__ p0, const float* __restrict__ p1,
    float4* __restrict__ xq_fw, float4* __restrict__ yq_fw,
    float4* __restrict__ xq_bw, float4* __restrict__ yq_bw, int n)
{
    int i = blockIdx.x * blockDim.x + threadIdx.x;
    if (i >= n) return;
    float p0x = p0[3*i+0], p0y = p0[3*i+1], p0z = p0[3*i+2];
    float p1x = p1[3*i+0], p1y = p1[3*i+1], p1z = p1[3*i+2];
    float fwx = fw[3*i+0], fwy = fw[3*i+1], fwz = fw[3*i+2];
    float bwx = bw[3*i+0], bwy = bw[3*i+1], bwz = bw[3*i+2];

    float fx = p0x + fwx, fy = p0y + fwy, fz = p0z + fwz;
    xq_fw[i] = make_float4(fx, fy, fz, fx*fx + fy*fy + fz*fz);

    float gx = fx - bwx, gy = fy - bwy, gz = fz - bwz;
    xq_bw[i] = make_float4(gx, gy, gz, gx*gx + gy*gy + gz*gz);

    yq_fw[i] = make_float4(p1x, p1y, p1z, p1x*p1x + p1y*p1y + p1z*p1z);
    yq_bw[i] = make_float4(p0x, p0y, p0z, p0x*p0x + p0y*p0y + p0z*p0z);
}

// ---------------------------------------------------------------------------
// Main kernel. Grid: (tilesPerSide/8, 2 sides, KSPLIT slices); block = 256
// (8 waves). Each wave owns 16 query points (WMMA N dim) and scans one
// uniform slice of target tiles (scalar trip count -> scalar loop + unroll).
// B (4x16 f32, per-wave constant): B[:,n] = (-2x0, -2x1, -2x2, 1).
// A (16x4 f32, per tile):          A[m,:] = ( y0,  y1,  y2, |y|^2).
// WMMA D[m,n] = |y_m|^2 - 2<x_n,y_m>; running min folded in VALU.
// ---------------------------------------------------------------------------
__global__ __launch_bounds__(256) void chamfer_min_wmma(
    const float4* __restrict__ xq_fw, const float4* __restrict__ yq_fw,
    const float4* __restrict__ xq_bw, const float4* __restrict__ yq_bw,
    float* __restrict__ dminP, int N, int Mfw, int Mbw)
{
    const int lane = threadIdx.x & 31;
    const int wib  = threadIdx.x >> 5;
    const int side  = blockIdx.y;                    // 0 = fw, 1 = bw (scalar)
    const int slice = blockIdx.z;                    // K slice (scalar)
    const int tile  = blockIdx.x * 8 + wib;          // query tile

    const float4* __restrict__ xq = side ? xq_bw : xq_fw;
    const float2* __restrict__ yq = (const float2*)(side ? yq_bw : yq_fw);
    const int kTiles        = (side ? Mbw : Mfw) >> 4;   // scalar
    const int tilesPerSlice = kTiles / KSPLIT;           // scalar
    const int kBeg          = slice * tilesPerSlice;     // scalar

    const int col = lane & 15;   // D-tile column n this lane serves
    const int hi  = lane >> 4;   // 0: K rows {0,1}, 1: K rows {2,3}

    // Fixed B operand for this wave's 16 query points.
    float4 xv = xq[tile * 16 + col];
    v2f B;
    B[0] = hi ? (-2.0f * xv.z) : (-2.0f * xv.x);
    B[1] = hi ? 1.0f           : (-2.0f * xv.y);

    float rm = 3.402823e38f;
    const v8f czero = {};

    // Lane reads float2 element (m*2 + hi) of the packed float4 array:
    // lane<16 -> (y0,y1), lane>=16 -> (y2,|y|^2) == the f32 16x4 A layout.
    // One global_load_b64 per lane per tile; stride 32 float2 per tile.
    const float2* __restrict__ yp = yq + ((size_t)(kBeg * 16 + col) * 2 + hi);

    float2 cur = yp[0];
    #pragma unroll 4
    for (int t = 0; t < tilesPerSlice - 1; ++t) {
        float2 nxt = yp[(size_t)(t + 1) * 32];   // prefetch next A tile
        v2f A; A[0] = cur.x; A[1] = cur.y;
        v8f d = __builtin_amdgcn_wmma_f32_16x16x4_f32(
            false, A, false, B, (short)0, czero, false, false);
        float m0 = fminf(fminf(d[0], d[1]), d[2]);
        float m1 = fminf(fminf(d[3], d[4]), d[5]);
        float m2 = fminf(d[6], d[7]);
        rm = fminf(rm, fminf(fminf(m0, m1), m2));
        cur = nxt;
    }
    {   // drain last tile
        v2f A; A[0] = cur.x; A[1] = cur.y;
        v8f d = __builtin_amdgcn_wmma_f32_16x16x4_f32(
            false, A, false, B, (short)0, czero, false, false);
        float m0 = fminf(fminf(d[0], d[1]), d[2]);
        float m1 = fminf(fminf(d[3], d[4]), d[5]);
        float m2 = fminf(d[6], d[7]);
        rm = fminf(rm, fminf(fminf(m0, m1), m2));
    }

    // Merge row halves: lane l has rows {0..7}, lane l+16 rows {8..15}.
    rm = fminf(rm, __shfl_xor(rm, 16, 32));

    // Per-slice raw partial min (epilogue deferred to finalize kernel).
    if (lane < 16)
        dminP[((size_t)side * KSPLIT + slice) * (size_t)N + tile * 16 + col] = rm;
}

// ---------------------------------------------------------------------------
// Finalize: min over slices, +|x|^2, clamp to [0, trunc], fixed-order sum.
// Single block -> bitwise deterministic.
// ---------------------------------------------------------------------------
__global__ __launch_bounds__(256) void chamfer_finalize(
    const float* __restrict__ dminP,
    const float4* __restrict__ xq_fw, const float4* __restrict__ xq_bw,
    float* __restrict__ out, int N)
{
    __shared__ float s[256];
    int t = threadIdx.x;
    float acc = 0.0f;
    for (int i = t; i < 2 * N; i += 256) {
        int side = (i >= N) ? 1 : 0;
        int j    = side ? (i - N) : i;
        const float* base = dminP + (size_t)side * KSPLIT * N + j;
        float m = base[0];
        #pragma unroll
        for (int sl = 1; sl < KSPLIT; ++sl) m = fminf(m, base[(size_t)sl * N]);
        float x2 = (side ? xq_bw : xq_fw)[j].w;
        acc += fminf(fmaxf(m + x2, 0.0f), TRUNC_CHAMFER);
    }
    s[t] = acc;
    __syncthreads();
    for (int k = 128; k > 0; k >>= 1) {
        if (t < k) s[t] += s[t + k];
        __syncthreads();
    }
    if (t == 0) out[0] = s[0] / (float)N;   // (sum_fw + sum_bw) / N
}

extern "C" void kernel_launch(void* const* d_in, const int* in_sizes, int n_in,
                              void* d_out, int out_size, void* d_ws, size_t ws_size,
                              hipStream_t stream) {
    const float* fw = (const float*)d_in[0];   // fw_flow_pred [N,3]
    const float* bw = (const float*)d_in[1];   // bw_flow_pred [N,3]
    const float* p0 = (const float*)d_in[2];   // pcl_0        [N,3]
    const float* p1 = (const float*)d_in[3];   // pcl_1        [M,3]

    const int N = in_sizes[2] / 3;             // 16384
    const int M = in_sizes[3] / 3;             // 16384

    float* ws = (float*)d_ws;
    float4* xq_fw = (float4*)(ws + (size_t)0 * 4 * N);
    float4* yq_fw = (float4*)(ws + (size_t)1 * 4 * N);
    float4* xq_bw = (float4*)(ws + (size_t)2 * 4 * N);
    float4* yq_bw = (float4*)(ws + (size_t)3 * 4 * N);
    float*  dminP = ws + (size_t)4 * 4 * N;    // [2][KSPLIT][N] partial mins

    chamfer_pack<<<(N + 255) / 256, 256, 0, stream>>>(
        fw, bw, p0, p1, xq_fw, yq_fw, xq_bw, yq_bw, N);

    dim3 grid((N / 16) / 8, 2, KSPLIT);        // (128, 2, 4) -> 8192 waves
    chamfer_min_wmma<<<grid, 256, 0, stream>>>(
        xq_fw, yq_fw, xq_bw, yq_bw, dminP, N, M, N);

    chamfer_finalize<<<1, 256, 0, stream>>>(
        dminP, xq_fw, xq_bw, (float*)d_out, N);
}